// TopKActivation_14577119002704
// MI455X (gfx1250) — compile-verified
//
#include <hip/hip_runtime.h>
#include <stdint.h>

#define NCOLS 16384
#define TPB   1024

#ifndef __has_builtin
#define __has_builtin(x) 0
#endif

#if __has_builtin(__builtin_amdgcn_global_load_async_to_lds_b128)
#define USE_ASYNC_LDS 1
#else
#define USE_ASYNC_LDS 0
#endif

// Vector pointee type the async-LDS builtin expects (from hipcc diagnostic):
// 'int __attribute__((__vector_size__(4 * sizeof(int))))'
typedef int v4i_b128 __attribute__((__vector_size__(16)));

// Monotonic float->uint key transform (order-preserving, bijective).
__device__ __forceinline__ uint32_t f2ord(uint32_t b) {
  uint32_t s = (uint32_t)((int32_t)b >> 31);       // 0 or 0xFFFFFFFF
  return b ^ (s | 0x80000000u);
}
__device__ __forceinline__ float ord2f(uint32_t u) {
  uint32_t b = (u & 0x80000000u) ? (u ^ 0x80000000u) : ~u;
  return __uint_as_float(b);
}

__global__ __launch_bounds__(TPB) void topk_scatter_kernel(
    const float* __restrict__ x, const int* __restrict__ kp,
    float* __restrict__ out) {
#if USE_ASYNC_LDS
  __shared__ uint32_t lds_row[NCOLS];              // 64 KB landing buffer
#endif
  __shared__ uint32_t hist[2048];                  // 8 KB radix histogram
  __shared__ uint32_t waveSum[32];
  __shared__ uint32_t s_b1, s_b12, s_tval, s_needeq, s_krem;

  const int t = threadIdx.x;
  const int row = blockIdx.x;
  const uint32_t k = (uint32_t)(*kp);
  const uint32_t* rowp = (const uint32_t*)(x + (size_t)row * NCOLS);

  if (t == 0) s_krem = k;

  uint32_t ukey[16];                               // this thread's 16 contiguous columns

#if USE_ASYNC_LDS
  {
    // Async DMA the whole row HBM -> LDS (16B per lane per op, 4 ops/thread).
    const char* gbase = (const char*)rowp;
    uint32_t lbase = (uint32_t)(uintptr_t)(void*)lds_row;  // low 32b = LDS byte offset
#pragma unroll
    for (int i = 0; i < 4; ++i) {
      uint32_t byteoff = (uint32_t)(t + i * TPB) * 16u;
      __builtin_amdgcn_global_load_async_to_lds_b128(
          (__attribute__((address_space(1))) v4i_b128*)(uintptr_t)(gbase + byteoff),
          (__attribute__((address_space(3))) v4i_b128*)(uint32_t)(lbase + byteoff),
          0, 0);
    }
#if __has_builtin(__builtin_amdgcn_s_wait_asynccnt)
    __builtin_amdgcn_s_wait_asynccnt(0);
#else
    asm volatile("s_wait_asynccnt 0" ::: "memory");
#endif
  }
  __syncthreads();
#pragma unroll
  for (int i = 0; i < 4; ++i) {
    uint4 v = ((const uint4*)lds_row)[t * 4 + i];  // columns [16t, 16t+16)
    ukey[4 * i + 0] = f2ord(v.x);
    ukey[4 * i + 1] = f2ord(v.y);
    ukey[4 * i + 2] = f2ord(v.z);
    ukey[4 * i + 3] = f2ord(v.w);
  }
#else
#pragma unroll
  for (int i = 0; i < 4; ++i) {
    uint4 v = ((const uint4*)rowp)[t * 4 + i];     // columns [16t, 16t+16)
    ukey[4 * i + 0] = f2ord(v.x);
    ukey[4 * i + 1] = f2ord(v.y);
    ukey[4 * i + 2] = f2ord(v.z);
    ukey[4 * i + 3] = f2ord(v.w);
  }
  __syncthreads();                                 // publish s_krem
#endif

  // ---- 3-level radix select (bits 31:21, 20:10, 9:0) for exact k-th value ----
  uint32_t b1 = 0, b12 = 0;
  for (int level = 0; level < 3; ++level) {
    hist[t] = 0u;
    hist[t + 1024] = 0u;
    __syncthreads();

    if (level == 0) {
#pragma unroll
      for (int i = 0; i < 16; ++i) atomicAdd(&hist[ukey[i] >> 21], 1u);
    } else if (level == 1) {
#pragma unroll
      for (int i = 0; i < 16; ++i)
        if ((ukey[i] >> 21) == b1) atomicAdd(&hist[(ukey[i] >> 10) & 0x7FFu], 1u);
    } else {
#pragma unroll
      for (int i = 0; i < 16; ++i)
        if ((ukey[i] >> 10) == b12) atomicAdd(&hist[ukey[i] & 0x3FFu], 1u);
    }
    __syncthreads();

    if (t < 32) {                                  // wave 0 does the descending select
      const int lane = t;
      const int nbins = (level == 2) ? 1024 : 2048;
      const int chunk = nbins >> 5;
      const int base = lane * chunk;
      const uint32_t kr = s_krem;
      uint32_t psum = 0;
      for (int i = 0; i < chunk; ++i) psum += hist[base + i];
      // inclusive suffix-sum across lanes (higher lane = higher bins)
      uint32_t r = psum;
#pragma unroll
      for (int d = 1; d < 32; d <<= 1) {
        uint32_t v = __shfl_down(r, d, 32);
        if (lane + d < 32) r += v;
      }
      uint32_t above = r - psum;                   // elements in bins above my chunk
      if (above < kr && r >= kr) {                 // exactly one lane wins
        uint32_t acc = above;
        int b = base;
        for (int i = chunk - 1; i >= 0; --i) {
          uint32_t h = hist[base + i];
          if (acc + h >= kr) { b = base + i; break; }
          acc += h;
        }
        if (level == 0)      s_b1 = (uint32_t)b;
        else if (level == 1) s_b12 = (s_b1 << 11) | (uint32_t)b;
        else { s_tval = (s_b12 << 10) | (uint32_t)b; s_needeq = kr - acc; }
        s_krem = kr - acc;
      }
    }
    __syncthreads();
    if (level == 0) b1 = s_b1;
    else if (level == 1) b12 = s_b12;
  }

  // ---- column-ordered ranking of elements == threshold (ties -> first occurrences) ----
  const uint32_t tv = s_tval;
  const uint32_t ne = s_needeq;
  uint32_t cnt = 0;
#pragma unroll
  for (int i = 0; i < 16; ++i) cnt += (ukey[i] == tv) ? 1u : 0u;

  const int lane = t & 31;
  const int wv = t >> 5;
  uint32_t incl = cnt;
#pragma unroll
  for (int d = 1; d < 32; d <<= 1) {
    uint32_t v = __shfl_up(incl, d, 32);
    if (lane >= d) incl += v;
  }
  if (lane == 31) waveSum[wv] = incl;
  __syncthreads();
  if (wv == 0) {                                   // 32 waves -> one wave scans wave totals
    uint32_t w = waveSum[lane];
    uint32_t wi = w;
#pragma unroll
    for (int d = 1; d < 32; d <<= 1) {
      uint32_t v = __shfl_up(wi, d, 32);
      if (lane >= d) wi += v;
    }
    waveSum[lane] = wi - w;                        // exclusive wave offset
  }
  __syncthreads();
  uint32_t rr = incl - cnt + waveSum[wv];          // my exclusive equal-rank base

  // ---- emit: x where (x > t) or (x == t and among first need_eq equals), else 0 ----
  float* orow = out + (size_t)row * NCOLS + (size_t)t * 16;
#pragma unroll
  for (int i = 0; i < 4; ++i) {
    float4 o;
    float* op = (float*)&o;
#pragma unroll
    for (int j = 0; j < 4; ++j) {
      uint32_t u = ukey[4 * i + j];
      bool keep;
      if (u > tv) keep = true;
      else if (u == tv) { keep = (rr < ne); rr++; }
      else keep = false;
      op[j] = keep ? ord2f(u) : 0.0f;
    }
    ((float4*)orow)[i] = o;
  }
}

extern "C" void kernel_launch(void* const* d_in, const int* in_sizes, int n_in,
                              void* d_out, int out_size, void* d_ws, size_t ws_size,
                              hipStream_t stream) {
  const float* x = (const float*)d_in[0];
  const int* kp = (const int*)d_in[1];
  float* out = (float*)d_out;
  unsigned rows = (unsigned)(in_sizes[0] / NCOLS);   // 8192
  topk_scatter_kernel<<<dim3(rows), dim3(TPB), 0, stream>>>(x, kp, out);
}